// RGCN_DistMult_84275848282312
// MI455X (gfx1250) — compile-verified
//
#include <hip/hip_runtime.h>

typedef __attribute__((ext_vector_type(2))) float v2f;
typedef __attribute__((ext_vector_type(8))) float v8f;

#define N_NODES   50000
#define N_RELS    32
#define DIM       128
#define N_EDGES   800000
#define N_TRIPLES 8192

// ---------------------------------------------------------------- zero int32
__global__ void zero_i32(int* __restrict__ p, int n) {
  int i = blockIdx.x * blockDim.x + threadIdx.x;
  if (i < n) p[i] = 0;
}

// ------------------------------------------------- per-(dst,rel) edge counts
__global__ void count_kernel(const int* __restrict__ dst,
                             const int* __restrict__ etype,
                             int* __restrict__ cnt) {
  int e = blockIdx.x * blockDim.x + threadIdx.x;
  if (e < N_EDGES) atomicAdd(&cnt[dst[e] * N_RELS + etype[e]], 1);
}

// --------------------------------------------------------- WMMA transforms
// For a 16-node tile: xb0 = X @ basis[0], xb1 = X @ basis[1],
// agg = X @ root + bias  (agg later receives the edge-message atomics).
// One workgroup = 16 nodes; 8 waves; wave w owns output columns [16w,16w+16).
// f32 WMMA fragment layout (ISA 7.12.2):
//   A 16x4: lane<16 -> row=lane, {v0,v1}={K0,K1}; lane>=16 -> {K2,K3}
//   B 4x16: v0 = rows K0(lanes 0-15) / K2(lanes 16-31); v1 = K1 / K3
//   C/D 16x16: vgpr v, lane<16 -> (row v, col lane); lane>=16 -> (row v+8)
__global__ __launch_bounds__(256) void transform_kernel(
    const float* __restrict__ x, const float* __restrict__ basis,
    const float* __restrict__ root, const float* __restrict__ bias,
    float* __restrict__ xb0, float* __restrict__ xb1, float* __restrict__ agg) {
  __shared__ float xs[16 * DIM];  // 8 KB X tile
  const int row0 = blockIdx.x * 16;

  // cooperative coalesced load: 2048 floats = 512 float4, 2 per thread
  {
    const float4* s4 = (const float4*)(x + row0 * DIM);
    float4* d4 = (float4*)xs;
    int t = threadIdx.x;
    d4[t] = s4[t];
    d4[t + 256] = s4[t + 256];
  }
  __syncthreads();

  const int wave = threadIdx.x >> 5;
  const int lane = threadIdx.x & 31;
  const int half = lane >> 4;   // 0: lanes 0-15, 1: lanes 16-31
  const int l16  = lane & 15;
  const int col0 = wave * 16;

  const float* __restrict__ W0 = basis;              // basis[0]: [K][N]
  const float* __restrict__ W1 = basis + DIM * DIM;  // basis[1]

  v8f acc0 = {}, acc1 = {}, accr = {};

#pragma unroll
  for (int k0 = 0; k0 < DIM; k0 += 4) {
    const int ka = k0 + 2 * half;
    v2f a;
    a[0] = xs[l16 * DIM + ka];
    a[1] = xs[l16 * DIM + ka + 1];
    const int i0 = ka * DIM + col0 + l16;
    const int i1 = i0 + DIM;
    v2f b0; b0[0] = W0[i0];   b0[1] = W0[i1];
    v2f b1; b1[0] = W1[i0];   b1[1] = W1[i1];
    v2f br; br[0] = root[i0]; br[1] = root[i1];
    acc0 = __builtin_amdgcn_wmma_f32_16x16x4_f32(false, a, false, b0,
                                                 (short)0, acc0, false, false);
    acc1 = __builtin_amdgcn_wmma_f32_16x16x4_f32(false, a, false, b1,
                                                 (short)0, acc1, false, false);
    accr = __builtin_amdgcn_wmma_f32_16x16x4_f32(false, a, false, br,
                                                 (short)0, accr, false, false);
  }

  const int col = col0 + l16;
  const float bv = bias[col];
#pragma unroll
  for (int v = 0; v < 8; v++) {
    const int row = row0 + v + 8 * half;
    xb0[row * DIM + col] = acc0[v];
    xb1[row * DIM + col] = acc1[v];
    agg[row * DIM + col] = accr[v] + bv;  // root term + bias seeds aggregation
  }
}

// ------------------------------------------------ per-edge message + scatter
// one wave32 per edge; lane covers 4 of the 128 dims via float4
__global__ __launch_bounds__(256) void edge_kernel(
    const float* __restrict__ xb0, const float* __restrict__ xb1,
    const int* __restrict__ src, const int* __restrict__ dst,
    const int* __restrict__ etype, const float* __restrict__ comp,
    const int* __restrict__ cnt, float* __restrict__ agg) {
  const int w = (blockIdx.x * blockDim.x + threadIdx.x) >> 5;
  const int lane = threadIdx.x & 31;
  if (w >= N_EDGES) return;
  const int s = src[w], d = dst[w], r = etype[w];
  const float norm = 1.0f / fmaxf((float)cnt[d * N_RELS + r], 1.0f);
  const float c0 = comp[r * 2 + 0] * norm;
  const float c1 = comp[r * 2 + 1] * norm;
  const float4 a = ((const float4*)(xb0 + s * DIM))[lane];
  const float4 b = ((const float4*)(xb1 + s * DIM))[lane];
  float* o = agg + d * DIM + lane * 4;
  atomicAdd(o + 0, c0 * a.x + c1 * b.x);
  atomicAdd(o + 1, c0 * a.y + c1 * b.y);
  atomicAdd(o + 2, c0 * a.z + c1 * b.z);
  atomicAdd(o + 3, c0 * a.w + c1 * b.w);
}

// ----------------------------------------------------------------- ReLU x4
__global__ void relu_kernel(float* __restrict__ p, int n4) {
  int i = blockIdx.x * blockDim.x + threadIdx.x;
  if (i < n4) {
    float4 v = ((float4*)p)[i];
    v.x = fmaxf(v.x, 0.f); v.y = fmaxf(v.y, 0.f);
    v.z = fmaxf(v.z, 0.f); v.w = fmaxf(v.w, 0.f);
    ((float4*)p)[i] = v;
  }
}

// ------------------------------------------------- DistMult: wave per triple
__global__ __launch_bounds__(256) void score_kernel(
    const float* __restrict__ x, const float* __restrict__ rel_emb,
    const int* __restrict__ h_idx, const int* __restrict__ r_idx,
    const int* __restrict__ t_idx, float* __restrict__ out, int n) {
  const int w = (blockIdx.x * blockDim.x + threadIdx.x) >> 5;
  const int lane = threadIdx.x & 31;
  if (w >= n) return;
  const int h = h_idx[w], r = r_idx[w], t = t_idx[w];
  const float4 a = ((const float4*)(x + h * DIM))[lane];
  const float4 b = ((const float4*)(rel_emb + r * DIM))[lane];
  const float4 c = ((const float4*)(x + t * DIM))[lane];
  float s = a.x * b.x * c.x + a.y * b.y * c.y +
            a.z * b.z * c.z + a.w * b.w * c.w;
#pragma unroll
  for (int off = 16; off > 0; off >>= 1) s += __shfl_down(s, off, 32);
  if (lane == 0) out[w] = s;
}

// ---------------------------------------------------------------- launcher
extern "C" void kernel_launch(void* const* d_in, const int* in_sizes, int n_in,
                              void* d_out, int out_size, void* d_ws, size_t ws_size,
                              hipStream_t stream) {
  const int* edge_index = (const int*)d_in[0];
  const int* src   = edge_index;             // edge_index[0]
  const int* dst   = edge_index + N_EDGES;   // edge_index[1]
  const int* etype = (const int*)d_in[1];
  const int* h_idx = (const int*)d_in[2];
  const int* r_idx = (const int*)d_in[3];
  const int* t_idx = (const int*)d_in[4];
  const float* node_emb = (const float*)d_in[5];
  const float* rel_emb  = (const float*)d_in[6];
  const float* basis1   = (const float*)d_in[7];
  const float* comp1    = (const float*)d_in[8];
  const float* root1    = (const float*)d_in[9];
  const float* bias1    = (const float*)d_in[10];
  const float* basis2   = (const float*)d_in[11];
  const float* comp2    = (const float*)d_in[12];
  const float* root2    = (const float*)d_in[13];
  const float* bias2    = (const float*)d_in[14];
  float* out = (float*)d_out;

  // workspace layout (all offsets 16B-aligned)
  char* ws = (char*)d_ws;
  int*   cnt  = (int*)(ws + 0);           //  6.4 MB: [N_NODES*N_RELS] i32
  float* xb0  = (float*)(ws + 6400000);   // 25.6 MB: [N_NODES][DIM]
  float* xb1  = (float*)(ws + 32000000);  // 25.6 MB
  float* agg1 = (float*)(ws + 57600000);  // 25.6 MB (doubles as x1 after ReLU)
  float* agg2 = (float*)(ws + 83200000);  // 25.6 MB (final node features)

  const int NTILES = N_NODES / 16;                 // 3125 (exact)
  const int EDGE_BLKS = (N_EDGES * 32) / 256;      // 100000 (exact)

  // per-(dst,rel) counts — identical for both layers, compute once
  zero_i32<<<(N_NODES * N_RELS + 255) / 256, 256, 0, stream>>>(cnt, N_NODES * N_RELS);
  count_kernel<<<(N_EDGES + 255) / 256, 256, 0, stream>>>(dst, etype, cnt);

  // layer 1
  transform_kernel<<<NTILES, 256, 0, stream>>>(node_emb, basis1, root1, bias1,
                                               xb0, xb1, agg1);
  edge_kernel<<<EDGE_BLKS, 256, 0, stream>>>(xb0, xb1, src, dst, etype,
                                             comp1, cnt, agg1);
  relu_kernel<<<(N_NODES * DIM / 4 + 255) / 256, 256, 0, stream>>>(agg1, N_NODES * DIM / 4);

  // layer 2
  transform_kernel<<<NTILES, 256, 0, stream>>>(agg1, basis2, root2, bias2,
                                               xb0, xb1, agg2);
  edge_kernel<<<EDGE_BLKS, 256, 0, stream>>>(xb0, xb1, src, dst, etype,
                                             comp2, cnt, agg2);

  // DistMult scores
  score_kernel<<<(N_TRIPLES * 32) / 256, 256, 0, stream>>>(agg2, rel_emb,
                                                           h_idx, r_idx, t_idx,
                                                           out, N_TRIPLES);
}